// Decompose_13477607375164
// MI455X (gfx1250) — compile-verified
//
#include <hip/hip_runtime.h>
#include <cstdint>

// ---------------------------------------------------------------------------
// Decompose: the reference collapses to a 5x5 scalar-tap stencil over
// clip(x,0,1) with reflect padding, + constant, /25, clip.  HBM-bound
// (~96 MB traffic => ~4.1 us floor @ 23.3 TB/s).  CDNA5 path used: Tensor
// Data Mover stages each (TH+4)x(TW+4) fp32 tile into LDS (TENSORcnt wait),
// VALU does the 25 FMAs/pixel from a sliding register window.
// ---------------------------------------------------------------------------

#define IMG_H 1024
#define IMG_W 1024
#define NPLANE 12          // bs*C = 4*3
#define TW 64              // output tile width  per block
#define TH 128             // output tile height per block
#define LW (TW + 4)        // LDS tile width  (68) -> row pitch, 272B = 17*16B
#define LH (TH + 4)        // LDS tile height (132)

typedef unsigned int tdm_v4u __attribute__((ext_vector_type(4)));
typedef int          tdm_v8i __attribute__((ext_vector_type(8)));
typedef int          tdm_v4i __attribute__((ext_vector_type(4)));

#if defined(__has_builtin)
#if __has_builtin(__builtin_amdgcn_tensor_load_to_lds) && __has_builtin(__builtin_amdgcn_s_wait_tensorcnt)
#define USE_TDM 1
#endif
#endif

// Fold the 25 E/S 1x1-conv pairs into 25 scalar taps (pre-divided by 25)
// plus one pixel-constant:  coef[0..24] = (wS_k . wE_k)/25,
//                           coef[25]    = (sum_k (wS_k . bE_k) + bS_k)/25
__global__ void coef_kernel(const float* __restrict__ wE, const float* __restrict__ bE,
                            const float* __restrict__ wS, const float* __restrict__ bS,
                            float* __restrict__ coef) {
    __shared__ float csum[25];
    const int k = threadIdx.x;
    if (k < 25) {
        float a = 0.0f, c = 0.0f;
        for (int d = 0; d < 64; ++d) {
            const float s = wS[k * 64 + d];
            a = fmaf(s, wE[k * 64 + d], a);
            c = fmaf(s, bE[k * 64 + d], c);
        }
        coef[k] = a * (1.0f / 25.0f);
        csum[k] = c + bS[k];
    }
    __syncthreads();
    if (k == 0) {
        float t = 0.0f;
        for (int i = 0; i < 25; ++i) t += csum[i];
        coef[25] = t * (1.0f / 25.0f);
    }
}

__device__ __forceinline__ int reflect_idx(int v, int n) {
    // numpy "reflect" (no edge repeat): -1->1, -2->2, n->n-2, n+1->n-3
    return v < 0 ? -v : (v >= n ? 2 * n - 2 - v : v);
}

__global__ __launch_bounds__(256) void stencil_kernel(const float* __restrict__ x,
                                                      const float* __restrict__ coef,
                                                      float* __restrict__ out) {
    __shared__ float tile[LH * LW];   // 35,904 B -> 8 blocks/WGP vs 320KB LDS

    const int bx = blockIdx.x, by = blockIdx.y, plane = blockIdx.z;
    const int tid = threadIdx.x;

    // Clamped load window: shifted inward at borders so every reflected index
    // the block needs lands inside [rlo,rlo+LH) x [clo,clo+LW).
    int rlo = by * TH - 2; rlo = rlo < 0 ? 0 : rlo; rlo = rlo > IMG_H - LH ? IMG_H - LH : rlo;
    int clo = bx * TW - 2; clo = clo < 0 ? 0 : clo; clo = clo > IMG_W - LW ? IMG_W - LW : clo;

    const float* src = x + ((size_t)plane * IMG_H + (size_t)rlo) * IMG_W + clo;

#ifdef USE_TDM
    if (tid < 32) {  // wave 0 issues one TDM 2D tile load (EXEC ignored by TDM)
        const uint64_t gaddr = (uint64_t)(uintptr_t)src;
        // Flat shared ptr lower 32 bits == wave-relative LDS byte offset (ISA 10.2)
        const unsigned ldsAddr = (unsigned)(uintptr_t)&tile[0];
        tdm_v4u g0;
        g0[0] = 1u;                                   // count=1, user descriptor
        g0[1] = ldsAddr;                              // lds_addr
        g0[2] = (unsigned)gaddr;                      // global_addr[31:0]
        g0[3] = (unsigned)((gaddr >> 32) & 0x01FFFFFFu) | 0x80000000u; // addr[56:32] | type=2
        tdm_v8i g1;
        g1[0] = 2 << 16;                              // data_size = 4B; mask=0; no pad/iter
        g1[1] = LW << 16;                             // tensor_dim0[15:0]
        g1[2] = LH << 16;                             // tensor_dim0 hi=0 | tensor_dim1[15:0]
        g1[3] = LW << 16;                             // tensor_dim1 hi=0 | tile_dim0=68
        g1[4] = LH;                                   // tile_dim1=132 | tile_dim2=0 (2D)
        g1[5] = IMG_W;                                // tensor_dim0_stride lo = 1024
        g1[6] = 0;                                    // stride0 hi | stride1 lo (unused 2D)
        g1[7] = 0;
        tdm_v4i g2 = {0, 0, 0, 0}, g3 = {0, 0, 0, 0}; // groups 2/3 unused (<=2D)
        tdm_v8i g4 = {0, 0, 0, 0, 0, 0, 0, 0};        // extended group (unused, zeros)
        __builtin_amdgcn_tensor_load_to_lds(g0, g1, g2, g3, g4, 0);
        __builtin_amdgcn_s_wait_tensorcnt(0);
    }
#else
    // Fallback: cooperative staging (keeps file compiling everywhere)
    for (int i = tid; i < LH * LW; i += 256) {
        const int r = i / LW, c = i - r * LW;
        tile[i] = src[(size_t)r * IMG_W + c];
    }
#endif
    __syncthreads();

    // clip(x,0,1) happens BEFORE padding in the reference -> clamp tile in place
    for (int i = tid; i < LH * LW; i += 256) {
        tile[i] = fminf(fmaxf(tile[i], 0.0f), 1.0f);
    }
    __syncthreads();

    // Taps (uniform addresses -> broadcast loads from L2-resident 104B buffer)
    float A[25];
#pragma unroll
    for (int k = 0; k < 25; ++k) A[k] = coef[k];
    const float cbar = coef[25];

    // Thread micro-tile: 4 wide x 8 tall.  16 threads across, 16 down.
    const int tx = tid & 15;
    const int ty = tid >> 4;
    const int c0 = bx * TW + tx * 4;   // 16B-aligned output column
    const int r0 = by * TH + ty * 8;

    // Per-thread column map (reflect once; identity for interior blocks)
    int lc[8];
#pragma unroll
    for (int j = 0; j < 8; ++j) lc[j] = reflect_idx(c0 - 2 + j, IMG_W) - clo;

    float win[5][8];   // sliding 5-row x 8-col register window
#pragma unroll
    for (int s = 0; s < 4; ++s) {      // prologue: rows r0-2 .. r0+1
        const int lr = reflect_idx(r0 - 2 + s, IMG_H) - rlo;
        const float* row = &tile[lr * LW];
#pragma unroll
        for (int j = 0; j < 8; ++j) win[s][j] = row[lc[j]];
    }

#pragma unroll
    for (int s = 0; s < 8; ++s) {
        const int lr = reflect_idx(r0 + 2 + s, IMG_H) - rlo;   // newest row
        {
            const float* row = &tile[lr * LW];
#pragma unroll
            for (int j = 0; j < 8; ++j) win[(s + 4) % 5][j] = row[lc[j]];
        }
        float acc0 = cbar, acc1 = cbar, acc2 = cbar, acc3 = cbar;
#pragma unroll
        for (int i = 0; i < 5; ++i) {
            const float* wr = win[(s + i) % 5];
#pragma unroll
            for (int j = 0; j < 5; ++j) {
                const float aij = A[i * 5 + j];
                acc0 = fmaf(aij, wr[j + 0], acc0);
                acc1 = fmaf(aij, wr[j + 1], acc1);
                acc2 = fmaf(aij, wr[j + 2], acc2);
                acc3 = fmaf(aij, wr[j + 3], acc3);
            }
        }
        float4 o;
        o.x = fminf(fmaxf(acc0, 0.0f), 1.0f);
        o.y = fminf(fmaxf(acc1, 0.0f), 1.0f);
        o.z = fminf(fmaxf(acc2, 0.0f), 1.0f);
        o.w = fminf(fmaxf(acc3, 0.0f), 1.0f);
        *reinterpret_cast<float4*>(out + ((size_t)plane * IMG_H + (r0 + s)) * IMG_W + c0) = o;
    }
}

extern "C" void kernel_launch(void* const* d_in, const int* in_sizes, int n_in,
                              void* d_out, int out_size, void* d_ws, size_t ws_size,
                              hipStream_t stream) {
    (void)in_sizes; (void)n_in; (void)out_size; (void)ws_size;
    const float* x  = (const float*)d_in[0];
    const float* wE = (const float*)d_in[1];
    const float* bE = (const float*)d_in[2];
    const float* wS = (const float*)d_in[3];
    const float* bS = (const float*)d_in[4];
    float* out  = (float*)d_out;
    float* coef = (float*)d_ws;   // 26 floats of scratch

    coef_kernel<<<1, 32, 0, stream>>>(wE, bE, wS, bS, coef);
    dim3 grid(IMG_W / TW, IMG_H / TH, NPLANE);   // 16 x 8 x 12 = 1536 blocks
    stencil_kernel<<<grid, 256, 0, stream>>>(x, coef, out);
}